// CapsShapeLayer_37709812859178
// MI455X (gfx1250) — compile-verified
//
#include <hip/hip_runtime.h>
#include <math.h>

// Problem: x [B,R,I], W [1,R,C,O,I] -> v [B,C,O]; 3 routing iterations.
#define Bdim 128
#define Rdim 1152
#define Cdim 16
#define Odim 16
#define Idim 8
#define COdim 256                 // C*O
#define RCHUNK 8                  // routes per s-pass block
#define NBLK (Rdim / RCHUNK)      // 144 K-split partials
#define INV_R (1.0f / 1152.0f)
#define INV_B (1.0f / 128.0f)

typedef __attribute__((ext_vector_type(2))) float v2f;
typedef __attribute__((ext_vector_type(8))) float v8f;

// WMMA f32 16x16x4 fragment model (codegen-verified in earlier rounds):
//  A (16x4): lane row M = lane&15; VGPR0/1 = K{2h, 2h+1}, h = lane>>4
//  B (4x16): lane col N = lane&15; VGPR0/1 = K rows {2h, 2h+1}
//  D (16x16): VGPR j -> M = j + 8h, N = lane&15

// -----------------------------------------------------------------------------
// S-pass: s[b,co] = sum_{r,i} (x[b,r,i] * c[r,cc]) * W[r,co,i],  cc == N-tile.
// u_hat is never materialized. Block = 8-route K-chunk; wave = batch M-tile.
// W slice (64KB) staged to LDS via global_load_async_to_lds_b128; x A-frags for
// the whole chunk live in 32 VGPRs. Branch-free: UNIFORM is a template param;
// weighted scales come from transposed cT[c][r] (contiguous, wave-uniform ->
// one wide s_load per N-tile). Writes f32 partials part[blk][b][co].
// -----------------------------------------------------------------------------
template <bool UNIFORM>
__global__ void __launch_bounds__(256)
caps_spass_wmma(const float* __restrict__ x, const float* __restrict__ W,
                const float* __restrict__ cmatT, float* __restrict__ part)
{
    __shared__ __align__(16) float wl[RCHUNK * COdim * Idim];   // 16384 f = 64KB

    const int blk     = blockIdx.x;          // 0..143
    const int r0      = blk * RCHUNK;
    const int wave    = threadIdx.x >> 5;    // M-tile (batch group)
    const int lane    = threadIdx.x & 31;
    const int lane_lo = lane & 15;
    const int h       = lane >> 4;

    // ---- stage W[r0..r0+7][*][*] into LDS via async copy (b128/lane) ----
    {
        const float* wg = W + (size_t)r0 * COdim * Idim;
        for (int off = threadIdx.x * 4; off < RCHUNK * COdim * Idim; off += 1024) {
            unsigned ldsb = (unsigned)(size_t)((__attribute__((address_space(3))) float*)wl + off);
            const float* ga = wg + off;
            asm volatile("global_load_async_to_lds_b128 %0, %1, off"
                         :: "v"(ldsb), "v"(ga) : "memory");
        }
        asm volatile("s_wait_asynccnt 0" ::: "memory");
    }

    // ---- preload x A-fragments for all 8 routes of the chunk (32 VGPRs) ----
    const int brow = wave * 16 + lane_lo;
    const float* xp = x + ((size_t)brow * Rdim + r0) * Idim;
    v2f A0[RCHUNK], A1[RCHUNK];
#pragma unroll
    for (int rr = 0; rr < RCHUNK; ++rr) {
        A0[rr] = *(const v2f*)(xp + rr * Idim + 2 * h);       // K = 0..3 of route
        A1[rr] = *(const v2f*)(xp + rr * Idim + 4 + 2 * h);   // K = 4..7
    }

    __syncthreads();   // LDS W slice visible to all waves

    const int bb = wave * 16 + h * 8;
    for (int n = 0; n < Cdim; ++n) {          // N-tile == capsule cc
        // wave-uniform scales for this capsule: 8 contiguous floats (32B-aligned)
        float csc[RCHUNK];
        if (!UNIFORM) {
            const float* cp = cmatT + n * Rdim + r0;
#pragma unroll
            for (int rr = 0; rr < RCHUNK; ++rr) csc[rr] = cp[rr];
        }

        v8f acc = {};
#pragma unroll
        for (int rr = 0; rr < RCHUNK; ++rr) {
            v2f a0 = A0[rr];
            v2f a1 = A1[rr];
            if (!UNIFORM) { a0 *= csc[rr]; a1 *= csc[rr]; }
            const float* lw = wl + ((rr * COdim + n * 16 + lane_lo) * Idim);
            const v2f b0 = *(const v2f*)(lw + 2 * h);
            const v2f b1 = *(const v2f*)(lw + 4 + 2 * h);
            acc = __builtin_amdgcn_wmma_f32_16x16x4_f32(false, a0, false, b0,
                                                        (short)0, acc, false, false);
            acc = __builtin_amdgcn_wmma_f32_16x16x4_f32(false, a1, false, b1,
                                                        (short)0, acc, false, false);
        }
        if (UNIFORM) {
#pragma unroll
            for (int j = 0; j < 8; ++j) acc[j] *= INV_R;   // post-scale: c == 1/R
        }
        // partials: part[blk][bb+j][n*16+lane_lo], co-contiguous across lanes
        float* outp = part + (((size_t)blk * Bdim + bb) * COdim + n * 16 + lane_lo);
#pragma unroll
        for (int j = 0; j < 8; ++j)
            outp[(size_t)j * COdim] = acc[j];
    }
}

// -----------------------------------------------------------------------------
// Reduce K-split partials and apply squash. Block = batch b, thread = co.
// -----------------------------------------------------------------------------
__global__ void __launch_bounds__(256)
caps_reduce_squash(const float* __restrict__ part, float* __restrict__ vout)
{
    const int b  = blockIdx.x;
    const int co = threadIdx.x;
    const float* p = part + (size_t)b * COdim + co;
    float s = 0.0f;
#pragma unroll 4
    for (int k = 0; k < NBLK; ++k)
        s += p[(size_t)k * Bdim * COdim];
    const float sn = s * s;
    vout[b * COdim + co] = sn * s / ((1.0f + sn) * sqrtf(sn));
}

// -----------------------------------------------------------------------------
// A-pass: a[r,cc] = (1/B) sum_{i,o} W[r,cc,o,i] * Q[r,i,cc,o],
//         Q[(r,i),(cc,o)] = sum_b x[b,r,i] * v[b,cc,o]   (K = B = 128 WMMA GEMM)
// One wave per (route-pair mtile, capsule). D-tile: r = mtile*2 + h, i = VGPR j,
// o = lane&15 -> fuse the W contraction as 8 FMAs/lane + shuffle reduce over o.
// -----------------------------------------------------------------------------
__global__ void __launch_bounds__(256)
caps_agree_wmma(const float* __restrict__ x, const float* __restrict__ W,
                const float* __restrict__ v, float* __restrict__ blog, int first)
{
    const int lane    = threadIdx.x & 31;
    const int lane_lo = lane & 15;
    const int h       = lane >> 4;

    const int job   = blockIdx.x * 8 + (threadIdx.x >> 5);   // 0..9215
    const int mtile = job >> 4;                              // route pair
    const int cc    = job & 15;                              // capsule

    // A[m,k] = x[b=k, r(m), i(m)]: r = mtile*2 + (m>>3), i = m&7
    const int ra = mtile * 2 + (lane_lo >> 3);
    const int ia = lane_lo & 7;
    const float* xa = x + (size_t)ra * Idim + ia;        // + k * R*I
    // B[k,n] = v[b=k, cc, o=n]
    const float* vb = v + cc * Odim + lane_lo;           // + k * 256

    v8f acc = {};
#pragma unroll 4
    for (int s = 0; s < 32; ++s) {
        const int k0 = 4 * s + 2 * h;
        v2f af, bf;
        af.x = xa[(size_t)k0 * (Rdim * Idim)];
        af.y = xa[(size_t)(k0 + 1) * (Rdim * Idim)];
        bf.x = vb[k0 * COdim];
        bf.y = vb[(k0 + 1) * COdim];
        acc = __builtin_amdgcn_wmma_f32_16x16x4_f32(false, af, false, bf,
                                                    (short)0, acc, false, false);
    }

    // acc[j] = Q[rq, i=j, cc, o=lane_lo], rq = mtile*2 + h
    const int rq = mtile * 2 + h;
    const float* wp = W + (((size_t)rq * Cdim + cc) * Odim + lane_lo) * Idim;
    float wsum = 0.0f;
#pragma unroll
    for (int j = 0; j < 8; ++j)
        wsum += wp[j] * acc[j];

    // reduce over o: lanes 0-15 and 16-31 separately (masks <= 8)
    for (int m = 1; m <= 8; m <<= 1)
        wsum += __shfl_xor(wsum, m, 32);

    if (lane_lo == 0) {
        const float a = wsum * INV_B;
        blog[rq * Cdim + cc] = first ? a : (blog[rq * Cdim + cc] + a);
    }
}

// -----------------------------------------------------------------------------
// softmax over routes axis of b[r,c]; one block per capsule column.
// Writes TRANSPOSED weights cT[c][r] (contiguous per capsule, coalesced here,
// and wide-s_load-able in the s-pass).
// -----------------------------------------------------------------------------
__global__ void __launch_bounds__(256)
caps_softmax_routes(const float* __restrict__ blog, float* __restrict__ cmatT)
{
    __shared__ float red[256];
    const int c = blockIdx.x;
    const int t = threadIdx.x;

    float m = -INFINITY;
    for (int r = t; r < Rdim; r += 256) m = fmaxf(m, blog[r * Cdim + c]);
    red[t] = m; __syncthreads();
    for (int s = 128; s > 0; s >>= 1) {
        if (t < s) red[t] = fmaxf(red[t], red[t + s]);
        __syncthreads();
    }
    const float mx = red[0];
    __syncthreads();

    float sm = 0.0f;
    for (int r = t; r < Rdim; r += 256) sm += expf(blog[r * Cdim + c] - mx);
    red[t] = sm; __syncthreads();
    for (int s = 128; s > 0; s >>= 1) {
        if (t < s) red[t] += red[t + s];
        __syncthreads();
    }
    const float inv = 1.0f / red[0];

    for (int r = t; r < Rdim; r += 256)
        cmatT[c * Rdim + r] = expf(blog[r * Cdim + c] - mx) * inv;
}

// -----------------------------------------------------------------------------
extern "C" void kernel_launch(void* const* d_in, const int* in_sizes, int n_in,
                              void* d_out, int out_size, void* d_ws, size_t ws_size,
                              hipStream_t stream)
{
    const float* x = (const float*)d_in[0];   // [128, 1152, 8]
    const float* W = (const float*)d_in[1];   // [1, 1152, 16, 16, 8]
    float* vfinal  = (float*)d_out;           // [128, 16, 16]

    // workspace (floats): partials | b_logits | cT | v   (~19.1 MB, L2-resident)
    float* part  = (float*)d_ws;                           // [NBLK, B, CO]
    float* blog  = part + (size_t)NBLK * Bdim * COdim;     // [R, C]
    float* cmatT = blog + Rdim * Cdim;                     // [C, R]
    float* vbuf  = cmatT + Rdim * Cdim;                    // [B, C, O]

    // ---- iteration 0: c uniform (softmax of zeros); b_logits = a ----
    caps_spass_wmma<true><<<NBLK, 256, 0, stream>>>(x, W, cmatT, part);
    caps_reduce_squash<<<Bdim, 256, 0, stream>>>(part, vbuf);
    caps_agree_wmma<<<Rdim, 256, 0, stream>>>(x, W, vbuf, blog, 1);

    // ---- iteration 1 ----
    caps_softmax_routes<<<Cdim, 256, 0, stream>>>(blog, cmatT);
    caps_spass_wmma<false><<<NBLK, 256, 0, stream>>>(x, W, cmatT, part);
    caps_reduce_squash<<<Bdim, 256, 0, stream>>>(part, vbuf);
    caps_agree_wmma<<<Rdim, 256, 0, stream>>>(x, W, vbuf, blog, 0);

    // ---- iteration 2 (final: squash straight to d_out) ----
    caps_softmax_routes<<<Cdim, 256, 0, stream>>>(blog, cmatT);
    caps_spass_wmma<false><<<NBLK, 256, 0, stream>>>(x, W, cmatT, part);
    caps_reduce_squash<<<Bdim, 256, 0, stream>>>(part, vfinal);
}